// Set2Set_58634893525277
// MI455X (gfx1250) — compile-verified
//
#include <hip/hip_runtime.h>
#include <hip/hip_bf16.h>
#include <math.h>

#define C      128
#define B      4096
#define NNODES 500000
#define NOUT   4
#define K_DIM  384   // 2C (q_star) + C (h0)
#define G_DIM  512   // 4C gates

typedef float v2f __attribute__((ext_vector_type(2)));
typedef float v8f __attribute__((ext_vector_type(8)));

__device__ __forceinline__ float sigf(float x) { return 1.f / (1.f + __expf(-x)); }

// X[:,0:256]=0 (q_star), X[:,256:384]=h, c0=h
__global__ void init_state(const float* __restrict__ h, float* __restrict__ X,
                           float* __restrict__ c0) {
    int idx = blockIdx.x * blockDim.x + threadIdx.x;   // B*K_DIM
    if (idx >= B * K_DIM) return;
    int b = idx / K_DIM, col = idx - b * K_DIM;
    if (col < 2 * C) {
        X[idx] = 0.f;
    } else {
        float hv = h[b * C + col - 2 * C];
        X[idx] = hv;
        c0[b * C + col - 2 * C] = hv;
    }
}

// Wt[k][g] = k<256 ? W_ih[g][k] : W_hh[g][k-256]   (shape [384,512])
__global__ void build_wt(const float* __restrict__ Wih, const float* __restrict__ Whh,
                         float* __restrict__ Wt) {
    int idx = blockIdx.x * blockDim.x + threadIdx.x;   // K_DIM*G_DIM
    if (idx >= K_DIM * G_DIM) return;
    int k = idx / G_DIM, g = idx - k * G_DIM;
    Wt[idx] = (k < 2 * C) ? Wih[g * (2 * C) + k] : Whh[g * C + (k - 2 * C)];
}

// seg[t] = lower_bound(batch, t); batch sorted ascending
__global__ void seg_bounds(const int* __restrict__ batch, int* __restrict__ seg) {
    int t = blockIdx.x * blockDim.x + threadIdx.x;
    if (t > B) return;
    int lo = 0, hi = NNODES;
    while (lo < hi) { int mid = (lo + hi) >> 1; if (batch[mid] < t) lo = mid + 1; else hi = mid; }
    seg[t] = lo;
}

// gates[B,512] = X[B,384] @ Wt[384,512] ; fp32 WMMA, 64x64 macro-tile, 8 waves
__global__ __launch_bounds__(256) void gemm_gates(const float* __restrict__ X,
                                                  const float* __restrict__ Wt,
                                                  float* __restrict__ gates) {
    __shared__ __align__(16) float Alds[64][20];  // pad 20: conflict-free 16-lane col reads, 16B-aligned f4 writes
    __shared__ __align__(16) float Blds[16][80];  // pad 80: half-wave rows land 32 banks apart

    const int tid  = threadIdx.x;
    const int wave = tid >> 5, lane = tid & 31;
    const int half = lane >> 4, lo = lane & 15;
    const int m_blk = blockIdx.x * 64, n_blk = blockIdx.y * 64;
    const int moff = (wave & 3) * 16;    // 0,16,32,48
    const int noff = (wave >> 2) * 32;   // 0,32
    const int ar = tid >> 2,  ac = (tid & 3)  * 4;   // A: 64x16 stage
    const int br = tid >> 4,  bc = (tid & 15) * 4;   // B: 16x64 stage

    v8f acc0 = {0.f,0.f,0.f,0.f,0.f,0.f,0.f,0.f};
    v8f acc1 = {0.f,0.f,0.f,0.f,0.f,0.f,0.f,0.f};

    for (int k0 = 0; k0 < K_DIM; k0 += 16) {
        const float4 a4 = *(const float4*)(X  + (size_t)(m_blk + ar) * K_DIM + k0 + ac);
        const float4 b4 = *(const float4*)(Wt + (size_t)(k0 + br)   * G_DIM + n_blk + bc);
        __syncthreads();
        *(float4*)&Alds[ar][ac] = a4;
        *(float4*)&Blds[br][bc] = b4;
        __syncthreads();
#pragma unroll
        for (int kk = 0; kk < 4; ++kk) {
            const int kc = kk * 4 + half * 2;   // lanes 0-15: K=4kk,4kk+1 ; lanes 16-31: +2,+3
            v2f a;  a.x  = Alds[moff + lo][kc];        a.y  = Alds[moff + lo][kc + 1];
            v2f b0; b0.x = Blds[kc][noff + lo];        b0.y = Blds[kc + 1][noff + lo];
            v2f b1; b1.x = Blds[kc][noff + 16 + lo];   b1.y = Blds[kc + 1][noff + 16 + lo];
            acc0 = __builtin_amdgcn_wmma_f32_16x16x4_f32(false, a, false, b0, (short)0, acc0, false, false);
            acc1 = __builtin_amdgcn_wmma_f32_16x16x4_f32(false, a, false, b1, (short)0, acc1, false, false);
        }
    }
    // C/D layout: VGPR r, lane(half,lo) -> [M = r + 8*half][N = lo]
    float* gp = gates + (size_t)(m_blk + moff + half * 8) * G_DIM + n_blk + noff + lo;
#pragma unroll
    for (int r = 0; r < 8; ++r) {
        gp[(size_t)r * G_DIM]      = acc0[r];
        gp[(size_t)r * G_DIM + 16] = acc1[r];
    }
}

// LSTM pointwise: c = sig(f)*c0 + sig(i)*tanh(g); h = sig(o)*tanh(c)
__global__ void lstm_pw(const float* __restrict__ gates, const float* __restrict__ bih,
                        const float* __restrict__ bhh, float* __restrict__ c0,
                        float* __restrict__ X) {
    int idx = blockIdx.x * blockDim.x + threadIdx.x;   // B*C
    if (idx >= B * C) return;
    int b = idx >> 7, j = idx & (C - 1);
    const float* g = gates + (size_t)b * G_DIM;
    float gi = g[j]         + bih[j]         + bhh[j];
    float gf = g[C + j]     + bih[C + j]     + bhh[C + j];
    float gg = g[2 * C + j] + bih[2 * C + j] + bhh[2 * C + j];
    float go = g[3 * C + j] + bih[3 * C + j] + bhh[3 * C + j];
    float c  = sigf(gf) * c0[idx] + sigf(gi) * tanhf(gg);
    float hn = sigf(go) * tanhf(c);
    c0[idx] = c;
    X[(size_t)b * K_DIM + j]         = hn;   // q_star.q
    X[(size_t)b * K_DIM + 2 * C + j] = hn;   // h0
}

// one workgroup per segment; 4 waves, single pass over k & v with online softmax
__global__ __launch_bounds__(128) void attn(const float* __restrict__ kmat,
                                            const float* __restrict__ vmat,
                                            float* X, const int* __restrict__ seg,
                                            float* __restrict__ out, int t) {
    const int b = blockIdx.x;
    const int tid = threadIdx.x;
    const int wave = tid >> 5, lane = tid & 31;
    __shared__ float r_l[4][C];
    __shared__ float m_l[4], s_l[4];

    const int n0 = seg[b], n1 = seg[b + 1];
    if (n0 >= n1) {   // empty segment -> r = 0
        out[(size_t)b * (NOUT * C) + t * C + tid] = 0.f;
        X[(size_t)b * K_DIM + C + tid] = 0.f;
        return;
    }
    const float4 q4 = *(const float4*)(X + (size_t)b * K_DIM + 2 * C + lane * 4);
    float m = -3.402823466e38f, s = 0.f;
    float4 r4 = make_float4(0.f, 0.f, 0.f, 0.f);

    for (int n = n0 + wave; n < n1; n += 4) {
        const float4 k4 = *(const float4*)(kmat + (size_t)n * C + lane * 4);
        float d = k4.x * q4.x + k4.y * q4.y + k4.z * q4.z + k4.w * q4.w;
        for (int o = 16; o; o >>= 1) d += __shfl_xor(d, o, 32);
        const float4 v4 = *(const float4*)(vmat + (size_t)n * C + lane * 4);
        float nm = fmaxf(m, d);
        float sc = __expf(m - nm);
        float p  = __expf(d - nm);
        r4.x = r4.x * sc + p * v4.x;
        r4.y = r4.y * sc + p * v4.y;
        r4.z = r4.z * sc + p * v4.z;
        r4.w = r4.w * sc + p * v4.w;
        s = s * sc + p;
        m = nm;
    }
    r_l[wave][lane * 4 + 0] = r4.x;
    r_l[wave][lane * 4 + 1] = r4.y;
    r_l[wave][lane * 4 + 2] = r4.z;
    r_l[wave][lane * 4 + 3] = r4.w;
    if (lane == 0) { m_l[wave] = m; s_l[wave] = s; }
    __syncthreads();

    float M = fmaxf(fmaxf(m_l[0], m_l[1]), fmaxf(m_l[2], m_l[3]));
    float S = 0.f, R = 0.f;
#pragma unroll
    for (int w = 0; w < 4; ++w) {
        float e = __expf(m_l[w] - M);
        S += s_l[w] * e;
        R += r_l[w][tid] * e;
    }
    float rv = R / (S + 1e-16f);
    out[(size_t)b * (NOUT * C) + t * C + tid] = rv;   // readout t
    X[(size_t)b * K_DIM + C + tid] = rv;              // q_star.r
}

extern "C" void kernel_launch(void* const* d_in, const int* in_sizes, int n_in,
                              void* d_out, int out_size, void* d_ws, size_t ws_size,
                              hipStream_t stream) {
    const float* kmat  = (const float*)d_in[0];
    const float* vmat  = (const float*)d_in[1];
    const float* h     = (const float*)d_in[2];
    const float* Wih   = (const float*)d_in[3];
    const float* Whh   = (const float*)d_in[4];
    const float* bih   = (const float*)d_in[5];
    const float* bhh   = (const float*)d_in[6];
    const int*   batch = (const int*)d_in[7];
    float* out = (float*)d_out;

    char* ws = (char*)d_ws;
    float* Wt    = (float*)(ws);                                    //  786,432 B
    float* X     = (float*)(ws + 786432);                           // 6,291,456 B
    float* gates = (float*)(ws + 786432 + 6291456);                 // 8,388,608 B
    float* c0    = (float*)(ws + 786432 + 6291456 + 8388608);       // 2,097,152 B
    int*   seg   = (int*)  (ws + 786432 + 6291456 + 8388608 + 2097152);

    init_state<<<(B * K_DIM + 255) / 256, 256, 0, stream>>>(h, X, c0);
    build_wt<<<(K_DIM * G_DIM + 255) / 256, 256, 0, stream>>>(Wih, Whh, Wt);
    seg_bounds<<<(B + 1 + 255) / 256, 256, 0, stream>>>(batch, seg);

    for (int t = 0; t < NOUT; ++t) {
        dim3 g(B / 64, G_DIM / 64);   // 64 x 8
        gemm_gates<<<g, 256, 0, stream>>>(X, Wt, gates);
        lstm_pw<<<(B * C + 255) / 256, 256, 0, stream>>>(gates, bih, bhh, c0, X);
        attn<<<B, 128, 0, stream>>>(kmat, vmat, X, seg, out, t);
    }
}